// masksampling_695784702485
// MI455X (gfx1250) — compile-verified
//
#include <hip/hip_runtime.h>

// Mask-sampling kernel for MI455X (gfx1250).
//
// Pure memory-bound op: ~32 MB streamed in, ~805 MB streamed out.
// Roofline floor at 23.3 TB/s HBM is ~36 us; no matrix math exists here, so
// no WMMA — the win comes from b128 non-temporal stores, a single NT b128
// load per 384 stored bytes, and global_prefetch for the streaming reads.

typedef int   __attribute__((ext_vector_type(4))) v4i;
typedef float __attribute__((ext_vector_type(4))) v4f;

namespace {

// PATTERNS packed: nibble index = pattern*2 + row; nibble = (col1<<2) | col0.
// P0=[[1,2],[0,0]] P1=[[0,1],[0,2]] P2=[[0,0],[2,1]] P3=[[2,0],[1,0]]
constexpr unsigned kLUT = 0x12608409u;

constexpr size_t   kN       = 32ull * 3ull * 1024ull * 1024ull; // elems per mask
constexpr unsigned kThreads = 1u << 20;                          // 4096 blocks * 256
constexpr int      kIters   = 2;                                 // 2^21 tiles total

// Two nibbles (4 column values) -> float4 of (value == tgt).
__device__ __forceinline__ v4f pack4(unsigned na, unsigned nb, unsigned tgt) {
    v4f f;
    f.x = ((na & 3u) == tgt) ? 1.0f : 0.0f;
    f.y = ((na >> 2) == tgt) ? 1.0f : 0.0f;
    f.z = ((nb & 3u) == tgt) ? 1.0f : 0.0f;
    f.w = ((nb >> 2) == tgt) ? 1.0f : 0.0f;
    return f;
}

} // namespace

__global__ __launch_bounds__(256) void masksampling_695784702485_kernel(
    const int* __restrict__ im, float* __restrict__ out) {
    const unsigned tid = blockIdx.x * 256u + threadIdx.x;
    const v4i* __restrict__ imv = (const v4i*)im;

#pragma unroll
    for (int it = 0; it < kIters; ++it) {
        const unsigned tile = tid + (unsigned)it * kThreads;

        // Stream-prefetch next iteration's initmask quad (gfx1250 global_prefetch_b8).
        if (it + 1 < kIters) {
            __builtin_prefetch((const void*)(imv + (tile + kThreads)), 0, 0);
        }

        // initmask flat index for this tile is exactly 4*tile -> one NT b128 load.
        const v4i q = __builtin_nontemporal_load(imv + tile);

        const unsigned hb = tile & 127u;          // h-block: 8 output columns
        const unsigned iw = (tile >> 7) & 511u;   // initmask row -> output rows 2*iw, 2*iw+1
        const unsigned b  = tile >> 16;           // batch

        const unsigned p0 = (unsigned)q.x & 3u;
        const unsigned p1 = (unsigned)q.y & 3u;
        const unsigned p2 = (unsigned)q.z & 3u;
        const unsigned p3 = (unsigned)q.w & 3u;

        // Row-0 and row-1 nibbles for each of the 4 pattern blocks.
        const unsigned n00 = (kLUT >> (p0 * 8u)) & 0xFu;
        const unsigned n01 = (kLUT >> (p1 * 8u)) & 0xFu;
        const unsigned n02 = (kLUT >> (p2 * 8u)) & 0xFu;
        const unsigned n03 = (kLUT >> (p3 * 8u)) & 0xFu;
        const unsigned n10 = (kLUT >> (p0 * 8u + 4u)) & 0xFu;
        const unsigned n11 = (kLUT >> (p1 * 8u + 4u)) & 0xFu;
        const unsigned n12 = (kLUT >> (p2 * 8u + 4u)) & 0xFu;
        const unsigned n13 = (kLUT >> (p3 * 8u + 4u)) & 0xFu;

        // 8 float4 tiles: {mask1,mask2} x {row0,row1} x {h[0..3],h[4..7]}
        const v4f m1r0a = pack4(n00, n01, 1u);
        const v4f m1r0b = pack4(n02, n03, 1u);
        const v4f m1r1a = pack4(n10, n11, 1u);
        const v4f m1r1b = pack4(n12, n13, 1u);
        const v4f m2r0a = pack4(n00, n01, 2u);
        const v4f m2r0b = pack4(n02, n03, 2u);
        const v4f m2r1a = pack4(n10, n11, 2u);
        const v4f m2r1b = pack4(n12, n13, 2u);

        // out[b][c][w][h]: base of (b, c=0, w=2*iw, h=8*hb). Channel stride 1<<20.
        const size_t base =
            (((size_t)b * 3u) << 20) + ((size_t)iw << 11) + ((size_t)hb << 3);
        float* __restrict__ o1 = out + base;        // mask1
        float* __restrict__ o2 = o1 + kN;           // mask2

#pragma unroll
        for (int c = 0; c < 3; ++c) {
            const size_t off = (size_t)c << 20;
            __builtin_nontemporal_store(m1r0a, (v4f*)(o1 + off));
            __builtin_nontemporal_store(m1r0b, (v4f*)(o1 + off + 4));
            __builtin_nontemporal_store(m1r1a, (v4f*)(o1 + off + 1024));
            __builtin_nontemporal_store(m1r1b, (v4f*)(o1 + off + 1028));
            __builtin_nontemporal_store(m2r0a, (v4f*)(o2 + off));
            __builtin_nontemporal_store(m2r0b, (v4f*)(o2 + off + 4));
            __builtin_nontemporal_store(m2r1a, (v4f*)(o2 + off + 1024));
            __builtin_nontemporal_store(m2r1b, (v4f*)(o2 + off + 1028));
        }
    }
}

extern "C" void kernel_launch(void* const* d_in, const int* in_sizes, int n_in,
                              void* d_out, int out_size, void* d_ws, size_t ws_size,
                              hipStream_t stream) {
    (void)in_sizes; (void)n_in; (void)out_size; (void)d_ws; (void)ws_size;
    // d_in[0] = x (float32, unused: masks depend only on initmask)
    // d_in[1] = initmask (integer, values 0..3)
    const int* initmask = (const int*)d_in[1];
    float*     out      = (float*)d_out;

    // 2^21 tiles, 2 per thread: 4096 blocks x 256 threads (8 waves/block, wave32).
    masksampling_695784702485_kernel<<<4096, 256, 0, stream>>>(initmask, out);
}